// GHMC_10273561772276
// MI455X (gfx1250) — compile-verified
//
#include <hip/hip_runtime.h>
#include <math.h>

// GHM-C loss, single streaming pass + tiny finisher.
// Memory-bound: 503 MB read once -> ~21.6 us floor at 23.3 TB/s.
// CDNA5 paths: ds_add_f32 LDS atomics (conflict-free [bin][tid] layout),
// v_wmma_f32_16x16x4_f32 deterministic 32-lane wave reduction,
// nontemporal global_load_b128 (read-once stream) + global_prefetch_b8.
// Hardware transcendentals (v_exp_f32/v_log_f32/v_rcp_f32) keep the loop
// off the VALU roofline; errors are ulp-scale on a 42M-element reduction.

#define BINS 10
#define TPB  256
#define NBLK 2048
#define NARR (2 * BINS)   // 10 weighted-bce sums + 10 valid counts

typedef float v2f   __attribute__((ext_vector_type(2)));
typedef float v8f   __attribute__((ext_vector_type(8)));
typedef float f32x4 __attribute__((ext_vector_type(4)));
typedef int   i32x4 __attribute__((ext_vector_type(4)));

// Deterministic sum of one float per lane across a wave32.
// A (16x4 f32, 2 VGPRs): vgpr0 = p, vgpr1 = 0; B = all-ones (layout-free).
// D[m][n] = p[m] + p[m+16]; lane0 holds M=0..7 in c[0..7], lane16 M=8..15.
// Requires EXEC all ones (call only from fully-converged code).
__device__ __forceinline__ float wave_sum32(float p) {
  v2f a; a[0] = p;    a[1] = 0.0f;
  v2f b; b[0] = 1.0f; b[1] = 1.0f;
  v8f c = {};
  c = __builtin_amdgcn_wmma_f32_16x16x4_f32(false, a, false, b,
                                            (short)0, c, false, false);
  float t = ((c[0] + c[1]) + (c[2] + c[3])) + ((c[4] + c[5]) + (c[6] + c[7]));
  int ti = __float_as_int(t);
  return __int_as_float(__builtin_amdgcn_readlane(ti, 0)) +
         __int_as_float(__builtin_amdgcn_readlane(ti, 16));
}

// One element: bin it, accumulate bce*weight and validity into the private
// LDS slots. hist layout: [array][tid] -> bank = tid mod 64, conflict-free
// regardless of bin divergence. ds_add_f32, in-order per wave -> deterministic.
__device__ __forceinline__ void ghm_elem(float p, int t, float w,
                                         float* hist, int tid) {
  float e   = __expf(-fabsf(p));                  // v_exp_f32
  float r   = __builtin_amdgcn_rcpf(1.0f + e);    // v_rcp_f32
  float sig = (p >= 0.0f) ? r : e * r;            // sigmoid(p)
  float tf  = (float)t;
  float g   = fabsf(sig - tf);                    // in [0,1]
  int bin   = (int)(g * (float)BINS);             // trunc == floor (g >= 0)
  if (bin > BINS - 1) bin = BINS - 1;
  float bce = fmaxf(p, 0.0f) - p * tf + __logf(1.0f + e);  // v_log_f32
  float valid = (w > 0.0f) ? 1.0f : 0.0f;
  atomicAdd(&hist[bin * TPB + tid], bce * w);                // ds_add_f32
  atomicAdd(&hist[BINS * TPB + bin * TPB + tid], valid);     // ds_add_f32
}

__global__ __launch_bounds__(TPB) void ghm_pass1(
    const f32x4* __restrict__ p4, const i32x4* __restrict__ t4,
    const f32x4* __restrict__ w4,
    const float* __restrict__ pred, const int* __restrict__ tgt,
    const float* __restrict__ wgt,
    float* __restrict__ partials, int nvec, int n) {
  __shared__ float hist[NARR * TPB];           // 20 KB -> 16 WGs/WGP by LDS
  const int tid = threadIdx.x;

  for (int k = tid; k < NARR * TPB; k += TPB) hist[k] = 0.0f;
  __syncthreads();

  const unsigned stride = gridDim.x * TPB;
  for (unsigned i = blockIdx.x * TPB + tid; i < (unsigned)nvec; i += stride) {
    f32x4 p = __builtin_nontemporal_load(&p4[i]);  // TH=NT: read-once stream
    i32x4 t = __builtin_nontemporal_load(&t4[i]);
    f32x4 w = __builtin_nontemporal_load(&w4[i]);
    __builtin_prefetch(&p4[i + stride], 0, 0);     // global_prefetch_b8
    __builtin_prefetch(&t4[i + stride], 0, 0);
    __builtin_prefetch(&w4[i + stride], 0, 0);
    ghm_elem(p[0], t[0], w[0], hist, tid);
    ghm_elem(p[1], t[1], w[1], hist, tid);
    ghm_elem(p[2], t[2], w[2], hist, tid);
    ghm_elem(p[3], t[3], w[3], hist, tid);
  }
  // scalar tail (N % 4), single thread; reconverges at the barrier below
  if (blockIdx.x == 0 && tid == 0) {
    for (int i = nvec * 4; i < n; ++i) ghm_elem(pred[i], tgt[i], wgt[i], hist, 0);
  }
  __syncthreads();

  // Block reduce: 8 waves cover the 20 arrays round-robin; fixed order,
  // fully converged per wave -> WMMA reduce is legal and deterministic.
  const int wave = tid >> 5, lane = tid & 31;
  for (int a = wave; a < NARR; a += TPB / 32) {
    float s = 0.0f;
#pragma unroll
    for (int r = 0; r < TPB; r += 32) s += hist[a * TPB + r + lane];
    float tot = wave_sum32(s);
    if (lane == 0) partials[a * gridDim.x + blockIdx.x] = tot;
  }
}

__global__ __launch_bounds__(640) void ghm_pass2(
    const float* __restrict__ partials, float* __restrict__ out, int nblk) {
  __shared__ float totals[NARR];
  const int tid = threadIdx.x, wave = tid >> 5, lane = tid & 31;

  // 20 waves, one array each; fixed-order lane accumulation + WMMA reduce.
  float s = 0.0f;
  for (int i = lane; i < nblk; i += 32) s += partials[wave * nblk + i];
  float tot = wave_sum32(s);
  if (lane == 0) totals[wave] = tot;
  __syncthreads();

  if (tid == 0) {
    float totv = 0.0f, nne = 0.0f;
    for (int b = 0; b < BINS; ++b) {
      float c = totals[BINS + b];
      totv += c;
      nne  += (c > 0.0f) ? 1.0f : 0.0f;
    }
    float tsum = fmaxf(totv, 1.0f);
    float nn   = fmaxf(nne, 1.0f);
    float loss = 0.0f;
    for (int b = 0; b < BINS; ++b) {
      float c = totals[BINS + b];
      float w = (c > 0.0f) ? (tsum / fmaxf(c, 1.0f)) : 0.0f;
      loss += (w / nn) * totals[b];
    }
    out[0] = (loss / tsum) * 1.0f;  // LOSS_WEIGHT = 1.0
  }
}

extern "C" void kernel_launch(void* const* d_in, const int* in_sizes, int n_in,
                              void* d_out, int out_size, void* d_ws, size_t ws_size,
                              hipStream_t stream) {
  const float* pred = (const float*)d_in[0];
  const int*   tgt  = (const int*)d_in[1];
  const float* wgt  = (const float*)d_in[2];
  const int n    = in_sizes[0];
  const int nvec = n >> 2;

  float* partials = (float*)d_ws;  // NARR * NBLK floats = 160 KB

  ghm_pass1<<<NBLK, TPB, 0, stream>>>(
      (const f32x4*)pred, (const i32x4*)tgt, (const f32x4*)wgt,
      pred, tgt, wgt, partials, nvec, n);
  ghm_pass2<<<1, 640, 0, stream>>>(partials, (float*)d_out, NBLK);
}